// GraphAttentionAggregator_6030134083937
// MI455X (gfx1250) — compile-verified
//
#include <hip/hip_runtime.h>

typedef float v2f __attribute__((ext_vector_type(2)));
typedef float v8f __attribute__((ext_vector_type(8)));

#define C_DIM 256   // feature dim
#define B_DIM 256   // batches
#define T_DIM 32    // virtual tokens

// ---------------------------------------------------------------------------
// Kernel 1: o[c] = sum_k W_out[c][k] * bv[k] + b_out[c]   (bv = b_in[2C:3C])
// via V_WMMA_F32_16X16X4_F32 (full f32 precision).
// One workgroup, 16 waves of 32; wave w computes output rows [16w, 16w+16).
//
// A (16x4 f32, documented layout): lanes 0-15 -> M=lane, V0=K0,V1=K1;
//                                  lanes 16-31 -> M=lane-16, V0=K2,V1=K3.
// B (4x16 f32, mirror layout): value depends only on K (bv replicated across
// all N columns), so every column of D holds the same partial o-tile.
// D (16x16 f32, documented): VGPR r: lanes 0-15 -> (M=r, N=lane);
//                                    lanes 16-31 -> (M=8+r, N=lane-16).
// ---------------------------------------------------------------------------
__global__ __launch_bounds__(512) void gaa_matvec_wmma(
    const float* __restrict__ Wout,   // (C, C) row-major
    const float* __restrict__ b_in,   // (3C,)
    const float* __restrict__ b_out,  // (C,)
    float* __restrict__ o)            // (C,) result in workspace
{
    const float* bv = b_in + 2 * C_DIM;

    const int lane    = threadIdx.x & 31;
    const int wave    = threadIdx.x >> 5;
    const int rowBase = wave * 16;
    const int m       = lane & 15;          // M within tile
    const int khalf   = (lane >> 4) << 1;   // 0 for lanes 0-15, 2 for 16-31
    const int row     = rowBase + m;

    v8f acc = {};
    #pragma unroll 4
    for (int kc = 0; kc < C_DIM; kc += 4) {
        const int k0 = kc + khalf;
        v2f a, b;
        a.x = Wout[row * C_DIM + k0];
        a.y = Wout[row * C_DIM + k0 + 1];
        b.x = bv[k0];
        b.y = bv[k0 + 1];
        // (neg_a, A, neg_b, B, c_mod, C, reuse_a, reuse_b)
        acc = __builtin_amdgcn_wmma_f32_16x16x4_f32(
            false, a, false, b, (short)0, acc, false, false);
    }

    // Every column N of D is identical; lanes with N==0 (lane 0 and lane 16)
    // write out their 8 rows of the tile.
    if ((lane & 15) == 0) {
        const int mBase = rowBase + ((lane >> 4) << 3);
        #pragma unroll
        for (int r = 0; r < 8; ++r) {
            const int out_row = mBase + r;
            o[out_row] = acc[r] + b_out[out_row];
        }
    }
}

// ---------------------------------------------------------------------------
// Kernel 2: broadcast o (256 f32, L2-hot) to d_out (B=256, T=32, C=256).
// Pure store-bandwidth: 8 MB of float4 stores (~0.36 us at 23.3 TB/s).
// ---------------------------------------------------------------------------
__global__ __launch_bounds__(256) void gaa_broadcast(
    const float4* __restrict__ o4,    // (C/4,) = 64 float4
    float4* __restrict__ out4)        // (B*T*C/4,)
{
    const int gid = blockIdx.x * blockDim.x + threadIdx.x;
    out4[gid] = o4[gid & 63];   // 64 float4 per 256-float row
}

extern "C" void kernel_launch(void* const* d_in, const int* in_sizes, int n_in,
                              void* d_out, int out_size, void* d_ws, size_t ws_size,
                              hipStream_t stream) {
    // setup_inputs order: x, virtual_tokens, W_in, b_in, W_out, b_out, batch
    const float* b_in  = (const float*)d_in[3];
    const float* Wout  = (const float*)d_in[4];
    const float* b_out = (const float*)d_in[5];

    float* o = (float*)d_ws;  // 256 floats of scratch

    gaa_matvec_wmma<<<1, 512, 0, stream>>>(Wout, b_in, b_out, o);

    const int total4 = (B_DIM * T_DIM * C_DIM) / 4;  // 524288 float4
    gaa_broadcast<<<total4 / 256, 256, 0, stream>>>(
        (const float4*)d_ws, (float4*)d_out);
}